// CalcImpute_5634997092783
// MI455X (gfx1250) — compile-verified
//
#include <hip/hip_runtime.h>

// KNN-impute per-row: top-16 smallest of 50000 distances, masked weighted mean
// of donors. Memory-bound: 1.64 GB single-pass stream -> ~70us at 23.3 TB/s.
//
// Strategy: O(1)-per-element streaming (running min + static-threshold candidate
// capture into LDS), provable threshold bound from the 16 smallest thread-minima,
// rare fallback rescan for arbitrary inputs, then exact (value,index) rank-select
// of the 16 winners and a reference-order 16-term reduction.
//
// Hot loop: two independent global_load_b128 (non-temporal) per iteration for
// MLP, one guard branch per 8 elements.

typedef float f4 __attribute__((ext_vector_type(4)));

#define NTHREADS 256
#define KSEL     16
#define CAP      4096
#define TPRED    2.0e-3f     // generous bound for uniform[0,1): E[#cand] ~ 100
#define SENTINEL 1.0e10f     // reference nan_to_num value

__device__ __forceinline__ float fixnan(float v) {
  return (v != v) ? SENTINEL : v;
}

__device__ __forceinline__ void append_cand(int* s_cnt, float* s_cv, int* s_ci,
                                            float v, int idx) {
  int p = atomicAdd(s_cnt, 1);
  if (p < CAP) { s_cv[p] = v; s_ci[p] = idx; }
}

__global__ __launch_bounds__(NTHREADS) void knn_impute_kernel(
    const float* __restrict__ dist,   // [rows, cols]
    const float* __restrict__ fitX,   // [cols]
    const int*   __restrict__ maskX,  // [cols], 1 = missing
    float*       __restrict__ out,    // [rows]
    int cols)
{
  __shared__ float s_minv[NTHREADS];
  __shared__ int   s_cnt;
  __shared__ int   s_thr_bits;       // float bits of threshold (all values >= 0)
  __shared__ float s_cv[CAP];
  __shared__ int   s_ci[CAP];
  __shared__ float s_num[KSEL];
  __shared__ float s_den[KSEL];

  const int tid = threadIdx.x;
  const int row = blockIdx.x;
  if (tid == 0) { s_cnt = 0; s_thr_bits = 0; }
  __syncthreads();

  const float* rowp = dist + (size_t)row * (size_t)cols;
  const f4* rp4 = (const f4*)rowp;   // row byte offset = row*cols*4 -> 16B aligned
  const int cols4 = cols >> 2;
  const int tail0 = cols4 << 2;

  // ---------------- Pass A: stream once, track min, capture candidates ----
  float mn = 3.0e38f;
  int i = tid;
  for (; i + NTHREADS < cols4; i += 2 * NTHREADS) {
    // two independent 128-bit NT loads -> memory clause, 2x outstanding loads
    f4 a = __builtin_nontemporal_load(rp4 + i);
    f4 b = __builtin_nontemporal_load(rp4 + i + NTHREADS);
    float a0 = fixnan(a.x), a1 = fixnan(a.y), a2 = fixnan(a.z), a3 = fixnan(a.w);
    float b0 = fixnan(b.x), b1 = fixnan(b.y), b2 = fixnan(b.z), b3 = fixnan(b.w);
    float m8 = fminf(fminf(fminf(a0, a1), fminf(a2, a3)),
                     fminf(fminf(b0, b1), fminf(b2, b3)));
    mn = fminf(mn, m8);
    if (m8 <= TPRED) {                           // rare: ~0.2% of elements qualify
      const int ba = i << 2;
      const int bb = (i + NTHREADS) << 2;
      if (a0 <= TPRED) append_cand(&s_cnt, s_cv, s_ci, a0, ba);
      if (a1 <= TPRED) append_cand(&s_cnt, s_cv, s_ci, a1, ba + 1);
      if (a2 <= TPRED) append_cand(&s_cnt, s_cv, s_ci, a2, ba + 2);
      if (a3 <= TPRED) append_cand(&s_cnt, s_cv, s_ci, a3, ba + 3);
      if (b0 <= TPRED) append_cand(&s_cnt, s_cv, s_ci, b0, bb);
      if (b1 <= TPRED) append_cand(&s_cnt, s_cv, s_ci, b1, bb + 1);
      if (b2 <= TPRED) append_cand(&s_cnt, s_cv, s_ci, b2, bb + 2);
      if (b3 <= TPRED) append_cand(&s_cnt, s_cv, s_ci, b3, bb + 3);
    }
  }
  for (; i < cols4; i += NTHREADS) {             // remainder f4 chunks
    f4 a = __builtin_nontemporal_load(rp4 + i);
    float a0 = fixnan(a.x), a1 = fixnan(a.y), a2 = fixnan(a.z), a3 = fixnan(a.w);
    float m4 = fminf(fminf(a0, a1), fminf(a2, a3));
    mn = fminf(mn, m4);
    if (m4 <= TPRED) {
      const int ba = i << 2;
      if (a0 <= TPRED) append_cand(&s_cnt, s_cv, s_ci, a0, ba);
      if (a1 <= TPRED) append_cand(&s_cnt, s_cv, s_ci, a1, ba + 1);
      if (a2 <= TPRED) append_cand(&s_cnt, s_cv, s_ci, a2, ba + 2);
      if (a3 <= TPRED) append_cand(&s_cnt, s_cv, s_ci, a3, ba + 3);
    }
  }
  for (int c = tail0 + tid; c < cols; c += NTHREADS) {   // scalar tail (cols%4)
    float x = fixnan(rowp[c]);
    mn = fminf(mn, x);
    if (x <= TPRED) append_cand(&s_cnt, s_cv, s_ci, x, c);
  }
  s_minv[tid] = mn;
  __syncthreads();

  // ---- threshold bound: 16th-smallest thread-minimum (>= true 16th value) --
  // The 16 smallest thread-minima are 16 distinct elements, so the true 16th
  // smallest element is <= the 16th smallest minimum.
  int rk = 0;
  for (int j = 0; j < NTHREADS; ++j)        // LDS same-address broadcast reads
    rk += (s_minv[j] < mn) ? 1 : 0;
  if (rk < KSEL)
    atomicMax(&s_thr_bits, __float_as_int(mn));  // int order == float order (>=0)
  __syncthreads();

  const float thr = __int_as_float(s_thr_bits);
  const int cnt0 = s_cnt;
  const bool need_rescan = (thr > TPRED) || (cnt0 > CAP);  // uniform across block
  __syncthreads();

  // ---------------- Fallback (exactness for arbitrary inputs) -------------
  if (need_rescan) {
    if (tid == 0) s_cnt = 0;
    __syncthreads();
    for (int k = tid; k < cols4; k += NTHREADS) {
      f4 a = rp4[k];                       // second pass, likely L2-warm
      float a0 = fixnan(a.x), a1 = fixnan(a.y), a2 = fixnan(a.z), a3 = fixnan(a.w);
      float m4 = fminf(fminf(a0, a1), fminf(a2, a3));
      if (m4 <= thr) {
        const int ba = k << 2;
        if (a0 <= thr) append_cand(&s_cnt, s_cv, s_ci, a0, ba);
        if (a1 <= thr) append_cand(&s_cnt, s_cv, s_ci, a1, ba + 1);
        if (a2 <= thr) append_cand(&s_cnt, s_cv, s_ci, a2, ba + 2);
        if (a3 <= thr) append_cand(&s_cnt, s_cv, s_ci, a3, ba + 3);
      }
    }
    for (int c = tail0 + tid; c < cols; c += NTHREADS) {
      float x = fixnan(rowp[c]);
      if (x <= thr) append_cand(&s_cnt, s_cv, s_ci, x, c);
    }
    __syncthreads();
  }

  const int cnt = min(s_cnt, CAP);   // provably >= 16 candidates collected

  // ------------- exact rank-select of top-16 by (value, index) ------------
  for (int t = tid; t < cnt; t += NTHREADS) {
    const float v  = s_cv[t];
    const int   ix = s_ci[t];
    int r = 0;
    for (int j = 0; j < cnt; ++j) {        // broadcast LDS reads
      const float ov = s_cv[j];
      const int   oi = s_ci[j];
      r += ((ov < v) || (ov == v && oi < ix)) ? 1 : 0;
    }
    if (r < KSEL) {                         // ranks unique (indices distinct)
      const float donor = fitX[ix];
      const float w     = (float)(1 - maskX[ix]);  // 1 = valid donor
      s_num[r] = donor * w;
      s_den[r] = w;
    }
  }
  __syncthreads();

  // ------------- reduce in reference order (ascending distance) -----------
  if (tid == 0) {
    float num = 0.0f, den = 0.0f;
    #pragma unroll
    for (int k = 0; k < KSEL; ++k) { num += s_num[k]; den += s_den[k]; }
    const float div = (den == 0.0f) ? 1.0f : den;
    out[row] = num / div;
  }
}

extern "C" void kernel_launch(void* const* d_in, const int* in_sizes, int n_in,
                              void* d_out, int out_size, void* d_ws, size_t ws_size,
                              hipStream_t stream) {
  // inputs: 0 = dist_pot_donors [R,C] f32, 1 = n_neighbors (scalar, ==16),
  //         2 = fit_X_col [C] f32, 3 = mask_fit_X_col [C] int
  const float* dist  = (const float*)d_in[0];
  const float* fitX  = (const float*)d_in[2];
  const int*   maskX = (const int*)d_in[3];
  float*       out   = (float*)d_out;

  const int cols = in_sizes[2];
  const int rows = in_sizes[0] / cols;

  knn_impute_kernel<<<rows, NTHREADS, 0, stream>>>(dist, fitX, maskX, out, cols);
}